// MultiBoxLoss_79594333929976
// MI455X (gfx1250) — compile-verified
//
#include <hip/hip_runtime.h>

#define B_ 64
#define P_ 16320
#define T_ 50
#define C_ 81

typedef __attribute__((ext_vector_type(2))) float v2f;
typedef __attribute__((ext_vector_type(8))) float v8f;

__device__ __forceinline__ unsigned int fkey(float v) {
    unsigned int u = __float_as_uint(v);
    return (v >= 0.0f) ? (u | 0x80000000u) : ~u;
}

// ---------------- K0: init scratch ----------------
__global__ void k_init(int* ovr, unsigned long long* truthKey, int* num_pos) {
    int i = blockIdx.x * blockDim.x + threadIdx.x;
    if (i < B_ * P_) ovr[i] = -1;
    if (i < B_ * T_) truthKey[i] = 0ull;
    if (i < B_) num_pos[i] = 0;
}

// ---------------- K1: jaccard match (per-prior best truth, per-truth best prior) ----------------
__global__ void k_match(const float* __restrict__ arm_loc, const float* __restrict__ arm_conf,
                        const float* __restrict__ priors, const float* __restrict__ targets,
                        float* __restrict__ best_ov, int* __restrict__ best_idx,
                        unsigned long long* __restrict__ truthKey) {
    __shared__ float tr[T_ * 4];
    __shared__ unsigned long long ldsKey[T_];
    int b = blockIdx.y;
    int tid = threadIdx.x;
    for (int i = tid; i < T_ * 4; i += blockDim.x) {
        int t = i >> 2, c = i & 3;
        tr[i] = targets[((size_t)b * T_ + t) * 5 + c];
    }
    if (tid < T_) ldsKey[tid] = 0ull;
    __syncthreads();

    int p = blockIdx.x * blockDim.x + tid;
    bool valid = (p < P_);
    float bx0 = 0.f, by0 = 0.f, bx1 = 0.f, by1 = 0.f, areaB = 0.f;
    bool ign = false;
    if (valid) {
        float pcx = priors[p * 4 + 0], pcy = priors[p * 4 + 1];
        float pw = priors[p * 4 + 2], ph = priors[p * 4 + 3];
        size_t base = ((size_t)b * P_ + p) * 4;
        float lx = arm_loc[base + 0], ly = arm_loc[base + 1];
        float lw = arm_loc[base + 2], lh = arm_loc[base + 3];
        float cx = pcx + lx * 0.1f * pw, cy = pcy + ly * 0.1f * ph;
        float w = pw * __expf(lw * 0.2f), h = ph * __expf(lh * 0.2f);
        bx0 = cx - w * 0.5f; by0 = cy - h * 0.5f;
        bx1 = cx + w * 0.5f; by1 = cy + h * 0.5f;
        areaB = (bx1 - bx0) * (by1 - by0);
        ign = arm_conf[((size_t)b * P_ + p) * 2 + 0] > 0.99f;
    }
    float bov = -1e30f;
    int bti = 0;
    int lane = tid & 31;
    for (int t = 0; t < T_; ++t) {
        float ax0 = tr[t * 4 + 0], ay0 = tr[t * 4 + 1];
        float ax1 = tr[t * 4 + 2], ay1 = tr[t * 4 + 3];
        unsigned long long key = 0ull;
        if (valid) {
            float ltx = fmaxf(ax0, bx0), lty = fmaxf(ay0, by0);
            float rbx = fminf(ax1, bx1), rby = fminf(ay1, by1);
            float iw = fmaxf(rbx - ltx, 0.0f), ih = fmaxf(rby - lty, 0.0f);
            float inter = iw * ih;
            float areaA = (ax1 - ax0) * (ay1 - ay0);
            float ov = inter / (areaA + areaB - inter);
            if (ign) ov = -1.0f;
            if (ov > bov) { bov = ov; bti = t; }
            key = ((unsigned long long)fkey(ov) << 32) |
                  (unsigned long long)(0xFFFFFFFFu - (unsigned)p);
        }
        // wave32 max-reduce of key
        #pragma unroll
        for (int o = 16; o > 0; o >>= 1) {
            unsigned long long other = __shfl_xor(key, o);
            if (other > key) key = other;
        }
        if (lane == 0) atomicMax(&ldsKey[t], key);
    }
    if (valid) {
        best_ov[(size_t)b * P_ + p] = bov;
        best_idx[(size_t)b * P_ + p] = bti;
    }
    __syncthreads();
    if (tid < T_) atomicMax(&truthKey[(size_t)b * T_ + tid], ldsKey[tid]);
}

// ---------------- K2: scatter overrides (last-t-wins via atomicMax) ----------------
__global__ void k_override(const unsigned long long* __restrict__ truthKey, int* __restrict__ ovr) {
    int b = blockIdx.x;
    int t = threadIdx.x;
    if (t < T_) {
        unsigned long long key = truthKey[(size_t)b * T_ + t];
        unsigned p = 0xFFFFFFFFu - (unsigned)(key & 0xFFFFFFFFull);
        atomicMax(&ovr[(size_t)b * P_ + (size_t)p], t);
    }
}

// ---------------- K3: finalize conf_t + localization loss ----------------
__global__ void k_encode(const float* __restrict__ arm_loc, const float* __restrict__ arm_conf,
                         const float* __restrict__ loc_data, const float* __restrict__ priors,
                         const float* __restrict__ targets,
                         const float* __restrict__ best_ov, const int* __restrict__ best_idx,
                         const int* __restrict__ ovr,
                         int* __restrict__ conf_t, int* __restrict__ num_pos,
                         float* __restrict__ partials_l) {
    __shared__ float red[256];
    __shared__ int cnt;
    int b = blockIdx.y;
    int tid = threadIdx.x;
    if (tid == 0) cnt = 0;
    __syncthreads();
    int p = blockIdx.x * blockDim.x + tid;
    float lsum = 0.0f;
    int posc = 0;
    if (p < P_) {
        float pcx = priors[p * 4 + 0], pcy = priors[p * 4 + 1];
        float pw = priors[p * 4 + 2], ph = priors[p * 4 + 3];
        size_t base4 = ((size_t)b * P_ + p) * 4;
        float lx = arm_loc[base4 + 0], ly = arm_loc[base4 + 1];
        float lw = arm_loc[base4 + 2], lh = arm_loc[base4 + 3];
        float cx = pcx + lx * 0.1f * pw, cy = pcy + ly * 0.1f * ph;
        float w = pw * __expf(lw * 0.2f), h = ph * __expf(lh * 0.2f);
        float bx0 = cx - w * 0.5f, by0 = cy - h * 0.5f;
        float bx1 = cx + w * 0.5f, by1 = cy + h * 0.5f;
        bool ign = arm_conf[((size_t)b * P_ + p) * 2 + 0] > 0.99f;

        size_t i = (size_t)b * P_ + p;
        float ov = best_ov[i];
        int idx = best_idx[i];
        int o = ovr[i];
        if (o >= 0) { idx = o; ov = 2.0f; }
        const float* tg = &targets[((size_t)b * T_ + idx) * 5];
        float mx0 = tg[0], my0 = tg[1], mx1 = tg[2], my1 = tg[3];
        int conf = (int)tg[4] + 1;
        if (ov < 0.5f) conf = 0;
        if (ign) conf = -1;
        conf_t[i] = conf;
        if (conf > 0) {
            posc = 1;
            float ccx = (bx0 + bx1) * 0.5f, ccy = (by0 + by1) * 0.5f;
            float cw = fmaxf(bx1 - bx0, 1e-8f), ch = fmaxf(by1 - by0, 1e-8f);
            float lt0 = ((mx0 + mx1) * 0.5f - ccx) / (0.1f * cw);
            float lt1 = ((my0 + my1) * 0.5f - ccy) / (0.1f * ch);
            float lt2 = __logf(fmaxf(mx1 - mx0, 1e-8f) / cw) / 0.2f;
            float lt3 = __logf(fmaxf(my1 - my0, 1e-8f) / ch) / 0.2f;
            float ltv[4] = { lt0, lt1, lt2, lt3 };
            #pragma unroll
            for (int c = 0; c < 4; ++c) {
                float d = loc_data[base4 + c] - ltv[c];
                float ad = fabsf(d);
                lsum += (ad < 1.0f) ? 0.5f * d * d : ad - 0.5f;
            }
        }
    }
    red[tid] = lsum;
    atomicAdd(&cnt, posc);
    __syncthreads();
    for (int s = 128; s > 0; s >>= 1) {
        if (tid < s) red[tid] += red[tid + s];
        __syncthreads();
    }
    if (tid == 0) {
        partials_l[blockIdx.y * gridDim.x + blockIdx.x] = red[0];
        atomicAdd(&num_pos[b], cnt);
    }
}

// ---------------- K4: cross-entropy via WMMA row-sum reduction ----------------
// Each wave handles 16 priors. Sum of exp over classes is computed as
// D = A(16x4 exp-tiles) x B(4x16 ones) accumulated across 21 WMMA ops (84 padded classes).
__global__ void k_ce(const float* __restrict__ conf_data, const int* __restrict__ conf_t,
                     float* __restrict__ ce, unsigned int* __restrict__ mkey) {
    int b = blockIdx.y;
    int tid = threadIdx.x;
    int wave = tid >> 5;
    int lane = tid & 31;
    int m = lane & 15;
    int half = lane >> 4;
    int base = blockIdx.x * 128 + wave * 16;
    if (base >= P_) return;   // wave-uniform exit; active waves keep EXEC all-ones for WMMA

    int prior = base + m;
    const float* row = conf_data + ((size_t)b * P_ + prior) * C_;
    int tgt = conf_t[(size_t)b * P_ + prior];
    int tcls = (tgt < 0) ? 0 : tgt;

    // pass 1: per-row max + picked logit (classes split between lane m and lane 16+m)
    float mx = -1e30f, pick = -1e30f;
    for (int k0 = 0; k0 < 84; k0 += 4) {
        int c0 = k0 + half * 2;
        #pragma unroll
        for (int j = 0; j < 2; ++j) {
            int c = c0 + j;
            if (c < C_) {
                float v = row[c];
                mx = fmaxf(mx, v);
                if (c == tcls) pick = v;
            }
        }
    }
    mx = fmaxf(mx, __shfl_xor(mx, 16));
    pick = fmaxf(pick, __shfl_xor(pick, 16));

    // pass 2: exp tiles -> WMMA accumulate (B = ones => D rows are row sums)
    v8f acc = { 0.f, 0.f, 0.f, 0.f, 0.f, 0.f, 0.f, 0.f };
    v2f ones;
    ones.x = 1.0f; ones.y = 1.0f;
    for (int k0 = 0; k0 < 84; k0 += 4) {
        int c0 = k0 + half * 2;
        v2f a;
        a.x = (c0 < C_) ? __expf(row[c0] - mx) : 0.0f;
        a.y = (c0 + 1 < C_) ? __expf(row[c0 + 1] - mx) : 0.0f;
        acc = __builtin_amdgcn_wmma_f32_16x16x4_f32(false, a, false, ones,
                                                    (short)0, acc, false, false);
    }
    // extract row sums: lane L<16 holds rows 0..7 in acc[0..7]; lane>=16 holds rows 8..15
    int r = m & 7;
    float rs = 0.0f;
    #pragma unroll
    for (int j = 0; j < 8; ++j) rs = (r == j) ? acc[j] : rs;
    int qrow = half * 8 + r;
    float mxq = __shfl(mx, qrow);
    float pickq = __shfl(pick, qrow);
    if (m < 8) {
        size_t q = (size_t)b * P_ + base + qrow;
        float c_e = mxq + __logf(rs) - pickq;   // lse - picked
        ce[q] = c_e;
        int ct = conf_t[q];
        float mine = (ct == 0) ? fmaxf(c_e, 0.0f) : 0.0f;  // pos or ignored -> 0
        mkey[q] = __float_as_uint(mine);                    // monotone bits (mine >= 0)
    }
}

// ---------------- K5: hard-negative mining (exact stable top-k via radix bisection) ----------------
__global__ void __launch_bounds__(1024) k_select(const unsigned int* __restrict__ mkey,
                                                 const int* __restrict__ num_pos,
                                                 int* __restrict__ neg_flag) {
    __shared__ unsigned int wred[32];
    __shared__ unsigned int woff[32];
    __shared__ unsigned int bcast;
    __shared__ unsigned int tieBase;
    int b = blockIdx.x;
    int tid = threadIdx.x;
    int lane = tid & 31, wv = tid >> 5;
    const unsigned int* key = mkey + (size_t)b * P_;
    int k = num_pos[b] * 3;
    if (k > P_ - 1) k = P_ - 1;
    if (k <= 0) {
        for (int p = tid; p < P_; p += 1024) neg_flag[(size_t)b * P_ + p] = 0;
        return;
    }
    // find k-th largest key value u: largest u with count(key >= u) >= k
    unsigned int u = 0;
    for (int bit = 31; bit >= 0; --bit) {
        unsigned int cand = u | (1u << bit);
        unsigned int cnt = 0;
        for (int p = tid; p < P_; p += 1024) cnt += (key[p] >= cand) ? 1u : 0u;
        #pragma unroll
        for (int o = 16; o > 0; o >>= 1) cnt += __shfl_xor(cnt, o);
        if (lane == 0) wred[wv] = cnt;
        __syncthreads();
        if (tid == 0) {
            unsigned int s = 0;
            for (int w = 0; w < 32; ++w) s += wred[w];
            bcast = s;
        }
        __syncthreads();
        if (bcast >= (unsigned int)k) u = cand;
        __syncthreads();
    }
    // count strictly greater
    unsigned int cg = 0;
    for (int p = tid; p < P_; p += 1024) cg += (key[p] > u) ? 1u : 0u;
    #pragma unroll
    for (int o = 16; o > 0; o >>= 1) cg += __shfl_xor(cg, o);
    if (lane == 0) wred[wv] = cg;
    __syncthreads();
    if (tid == 0) {
        unsigned int s = 0;
        for (int w = 0; w < 32; ++w) s += wred[w];
        bcast = s;
        tieBase = 0;
    }
    __syncthreads();
    unsigned int rem = (unsigned int)k - bcast;   // ties to take, lowest index first
    __syncthreads();
    for (int chunk = 0; chunk < P_; chunk += 1024) {
        int p = chunk + tid;
        bool inb = p < P_;
        unsigned int kv = inb ? key[p] : 0u;
        bool tie = inb && (kv == u);
        unsigned long long mask = __ballot(tie);                 // wave32: low 32 bits
        unsigned int lanePre = (unsigned int)__popcll(mask & ((1ull << lane) - 1ull));
        unsigned int wcnt = (unsigned int)__popcll(mask);
        if (lane == 0) wred[wv] = wcnt;
        __syncthreads();
        if (tid == 0) {
            unsigned int run = tieBase;
            for (int w = 0; w < 32; ++w) { woff[w] = run; run += wred[w]; }
            tieBase = run;
        }
        __syncthreads();
        if (inb) {
            unsigned int rank = woff[wv] + lanePre;
            int ng = tie ? ((rank < rem) ? 1 : 0) : ((kv > u) ? 1 : 0);
            neg_flag[(size_t)b * P_ + p] = ng;
        }
        __syncthreads();
    }
}

// ---------------- K6: confidence loss partials ----------------
__global__ void k_conf_loss(const float* __restrict__ ce, const int* __restrict__ conf_t,
                            const int* __restrict__ neg_flag, float* __restrict__ partials_c) {
    __shared__ float red[256];
    int b = blockIdx.y;
    int tid = threadIdx.x;
    int p = blockIdx.x * 256 + tid;
    float v = 0.0f;
    if (p < P_) {
        size_t i = (size_t)b * P_ + p;
        if (conf_t[i] > 0 || neg_flag[i]) v = ce[i];
    }
    red[tid] = v;
    __syncthreads();
    for (int s = 128; s > 0; s >>= 1) {
        if (tid < s) red[tid] += red[tid + s];
        __syncthreads();
    }
    if (tid == 0) partials_c[blockIdx.y * gridDim.x + blockIdx.x] = red[0];
}

// ---------------- K7: deterministic final reduction ----------------
__global__ void k_final(const float* __restrict__ partials_l, const float* __restrict__ partials_c,
                        const int* __restrict__ num_pos, float* __restrict__ out, int nparts) {
    __shared__ float redl[256], redc[256];
    __shared__ int redn[256];
    int tid = threadIdx.x;
    float sl = 0.0f, sc = 0.0f;
    int sn = 0;
    for (int i = tid; i < nparts; i += 256) { sl += partials_l[i]; sc += partials_c[i]; }
    for (int i = tid; i < B_; i += 256) sn += num_pos[i];
    redl[tid] = sl; redc[tid] = sc; redn[tid] = sn;
    __syncthreads();
    for (int s = 128; s > 0; s >>= 1) {
        if (tid < s) {
            redl[tid] += redl[tid + s];
            redc[tid] += redc[tid + s];
            redn[tid] += redn[tid + s];
        }
        __syncthreads();
    }
    if (tid == 0) {
        float N = (float)redn[0];
        out[0] = redl[0] / N;
        out[1] = redc[0] / N;
    }
}

extern "C" void kernel_launch(void* const* d_in, const int* in_sizes, int n_in,
                              void* d_out, int out_size, void* d_ws, size_t ws_size,
                              hipStream_t stream) {
    (void)in_sizes; (void)n_in; (void)out_size; (void)ws_size;
    const float* arm_loc   = (const float*)d_in[0];
    const float* arm_conf  = (const float*)d_in[1];
    const float* loc_data  = (const float*)d_in[2];
    const float* conf_data = (const float*)d_in[3];
    const float* priors    = (const float*)d_in[4];
    const float* targets   = (const float*)d_in[5];
    float* out = (float*)d_out;

    char* w = (char*)d_ws;
    size_t BP = (size_t)B_ * P_;
    unsigned long long* truthKey = (unsigned long long*)(w);             // 25600 B
    int*   ovr        = (int*)  (w + 25600);
    float* best_ov    = (float*)(w + 25600 + BP * 4);
    int*   best_idx   = (int*)  (w + 25600 + BP * 8);
    int*   conf_t     = (int*)  (w + 25600 + BP * 12);
    float* ce         = (float*)(w + 25600 + BP * 16);
    unsigned int* mkey = (unsigned int*)(w + 25600 + BP * 20);
    int*   neg_flag   = (int*)  (w + 25600 + BP * 24);
    int*   num_pos    = (int*)  (w + 25600 + BP * 28);
    float* partials_l = (float*)(w + 25600 + BP * 28 + 256);
    float* partials_c = (float*)(w + 25600 + BP * 28 + 256 + 4096 * 4);

    int pb = (P_ + 255) / 256;  // 64 blocks per batch

    k_init<<<dim3((unsigned)((BP + 255) / 256)), dim3(256), 0, stream>>>(ovr, truthKey, num_pos);
    k_match<<<dim3(pb, B_), dim3(256), 0, stream>>>(arm_loc, arm_conf, priors, targets,
                                                    best_ov, best_idx, truthKey);
    k_override<<<dim3(B_), dim3(64), 0, stream>>>(truthKey, ovr);
    k_encode<<<dim3(pb, B_), dim3(256), 0, stream>>>(arm_loc, arm_conf, loc_data, priors, targets,
                                                     best_ov, best_idx, ovr,
                                                     conf_t, num_pos, partials_l);
    k_ce<<<dim3(128, B_), dim3(256), 0, stream>>>(conf_data, conf_t, ce, mkey);
    k_select<<<dim3(B_), dim3(1024), 0, stream>>>(mkey, num_pos, neg_flag);
    k_conf_loss<<<dim3(pb, B_), dim3(256), 0, stream>>>(ce, conf_t, neg_flag, partials_c);
    k_final<<<dim3(1), dim3(256), 0, stream>>>(partials_l, partials_c, num_pos, out, pb * B_);
}